// RGCN_28432683499969
// MI455X (gfx1250) — compile-verified
//
#include <hip/hip_runtime.h>

typedef __attribute__((ext_vector_type(2))) float v2f;
typedef __attribute__((ext_vector_type(8))) float v8f;

#define N_NODES 50000
#define N_EDGES 800000
#define HDIM    160
#define LEAK    0.01f
#define ROWTILES 3125          // 50000 / 16, exact
#define KPAD    780            // even (float2-aligned); 12c mod 64 distinct -> bank-friendly

// ---------------------------------------------------------------------------
// fp32 WMMA GEMM, guard-free fast path.
// Requirements: M % 16 == 0, K % 16 == 0, Nc % 16 == 0, lda == K.
// Block = 4 waves; block computes rows [64*blockIdx.x .. +64) x 16 cols.
// B tile (K x 16) staged in LDS transposed; shared by all 4 waves.
// Two independent accumulator chains to keep the WMMA pipe busy.
// ---------------------------------------------------------------------------
__global__ __launch_bounds__(128) void gemm_wmma_f32(
    const float* __restrict__ A,            // M x K row-major
    const float* __restrict__ W, int ldw,   // K x (>= col0+16) row-major
    const float* __restrict__ bias,         // may be nullptr
    float* __restrict__ C, int ldc,
    int K, int apply_lrelu)
{
    __shared__ float ldsB[16 * KPAD];

    const int tid  = threadIdx.x;
    const int col0 = blockIdx.y * 16;

    // Cooperative fill: ldsB[col][k] = W[k][col0+col]
    for (int idx = tid; idx < (K << 4); idx += 128) {
        const int k   = idx >> 4;
        const int col = idx & 15;
        ldsB[col * KPAD + k] = W[(long)k * ldw + col0 + col];
    }
    __syncthreads();

    const int wave    = tid >> 5;
    const int lane    = tid & 31;
    const int half    = lane >> 4;          // K sub-pair selector
    const int l16     = lane & 15;
    const int rowTile = blockIdx.x * 4 + wave;
    if (rowTile >= ROWTILES) return;        // after barrier: safe

    const int row0 = rowTile * 16;
    const float* __restrict__ ap = A + (long)(row0 + l16) * K + half * 2;
    const float* __restrict__ bp = ldsB + l16 * KPAD + half * 2;

    v8f c0 = {};
    v8f c1 = {};
    for (int k0 = 0; k0 < K; k0 += 16) {
        __builtin_prefetch(ap + k0 + 16, 0, 3);        // WGP-scope prefetch, next chunk
        const v2f a0 = *(const v2f*)(ap + k0);
        const v2f a1 = *(const v2f*)(ap + k0 + 4);
        const v2f a2 = *(const v2f*)(ap + k0 + 8);
        const v2f a3 = *(const v2f*)(ap + k0 + 12);
        const v2f b0 = *(const v2f*)(bp + k0);
        const v2f b1 = *(const v2f*)(bp + k0 + 4);
        const v2f b2 = *(const v2f*)(bp + k0 + 8);
        const v2f b3 = *(const v2f*)(bp + k0 + 12);
        c0 = __builtin_amdgcn_wmma_f32_16x16x4_f32(false, a0, false, b0, (short)0, c0, false, false);
        c1 = __builtin_amdgcn_wmma_f32_16x16x4_f32(false, a1, false, b1, (short)0, c1, false, false);
        c0 = __builtin_amdgcn_wmma_f32_16x16x4_f32(false, a2, false, b2, (short)0, c0, false, false);
        c1 = __builtin_amdgcn_wmma_f32_16x16x4_f32(false, a3, false, b3, (short)0, c1, false, false);
    }
    const v8f c = c0 + c1;

    const int   col = col0 + l16;
    const float bv  = bias ? bias[col] : 0.0f;

#pragma unroll
    for (int v = 0; v < 8; ++v) {
        const int row = row0 + v + half * 8;   // C/D layout: VGPR v -> rows v / v+8
        float val = c[v] + bv;
        if (apply_lrelu) val = (val >= 0.0f) ? val : LEAK * val;
        C[(long)row * ldc + col] = val;
    }
}

// ---------------------------------------------------------------------------
// Tiny ragged encoders (K = 6 or 11): one thread per (node, out-col of 32).
// ---------------------------------------------------------------------------
__global__ void encode_small(const float* __restrict__ X, int K,
                             const float* __restrict__ W,   // K x 32
                             const float* __restrict__ b,
                             float* __restrict__ C)         // h + colOff, ldc=HDIM
{
    int idx = blockIdx.x * blockDim.x + threadIdx.x;
    if (idx >= N_NODES * 32) return;
    const int n   = idx >> 5;
    const int col = idx & 31;
    float s = b[col];
    const float* x = X + (long)n * K;
    for (int k = 0; k < K; ++k) s += x[k] * W[k * 32 + col];
    s = (s >= 0.0f) ? s : LEAK * s;
    C[(long)n * HDIM + col] = s;
}

// ---------------------------------------------------------------------------
// Per-relation mean aggregation: thread = (edge, 4-feature group).
// ---------------------------------------------------------------------------
__global__ void scatter_rel4(const float* __restrict__ t,
                             const int* __restrict__ src,
                             const int* __restrict__ dst,
                             const int* __restrict__ etype,
                             int rel,
                             float* __restrict__ acc,
                             float* __restrict__ cnt)
{
    long gid = (long)blockIdx.x * blockDim.x + threadIdx.x;
    const long total = (long)N_EDGES * (HDIM / 4);
    if (gid >= total) return;
    const int e = (int)(gid / (HDIM / 4));
    const int g = (int)(gid - (long)e * (HDIM / 4));
    if (etype[e] != rel) return;
    const int s = src[e];
    const int d = dst[e];
    const int col = g * 4;
    const float4 v = *(const float4*)(t + (long)s * HDIM + col);
    float* a = acc + (long)d * HDIM + col;
    atomicAdd(a + 0, v.x);
    atomicAdd(a + 1, v.y);
    atomicAdd(a + 2, v.z);
    atomicAdd(a + 3, v.w);
    if (g == 0) atomicAdd(&cnt[d], 1.0f);
}

__global__ void combine_mean4(float* __restrict__ out,
                              const float* __restrict__ acc,
                              const float* __restrict__ cnt)
{
    long gid = (long)blockIdx.x * blockDim.x + threadIdx.x;
    const long total = (long)N_NODES * (HDIM / 4);
    if (gid >= total) return;
    const int node = (int)(gid / (HDIM / 4));
    float c = cnt[node];
    if (c < 1.0f) c = 1.0f;
    const float inv = 1.0f / c;
    const float4 av = *(const float4*)(acc + gid * 4);
    float4* ov = (float4*)(out + gid * 4);
    float4 o = *ov;
    o.x += av.x * inv; o.y += av.y * inv; o.z += av.z * inv; o.w += av.w * inv;
    *ov = o;
}

// Final classifier head: out[i,0:2] = em[i,:] @ W_o2 (80x2) + b_o2
__global__ void out_head(const float* __restrict__ em,
                         const float* __restrict__ W,
                         const float* __restrict__ b,
                         float* __restrict__ out)
{
    int i = blockIdx.x * blockDim.x + threadIdx.x;
    if (i >= N_NODES) return;
    float a0 = b[0], a1 = b[1];
    const float* e = em + (long)i * 80;
#pragma unroll 8
    for (int k = 0; k < 80; ++k) {
        const float v = e[k];
        a0 += v * W[k * 2 + 0];
        a1 += v * W[k * 2 + 1];
    }
    out[(long)i * 2 + 0] = a0;
    out[(long)i * 2 + 1] = a1;
}

// ---------------------------------------------------------------------------
extern "C" void kernel_launch(void* const* d_in, const int* in_sizes, int n_in,
                              void* d_out, int out_size, void* d_ws, size_t ws_size,
                              hipStream_t stream)
{
    (void)in_sizes; (void)n_in; (void)out_size; (void)ws_size;

    const float* pre_x   = (const float*)d_in[0];
    const int*   eidx    = (const int*)d_in[2];
    const int*   etype   = (const int*)d_in[3];
    const float* nprop   = (const float*)d_in[4];
    const float* ncat    = (const float*)d_in[5];
    const float* des     = (const float*)d_in[6];
    const float* tweet   = (const float*)d_in[7];
    const float* W_np    = (const float*)d_in[8];
    const float* b_np    = (const float*)d_in[9];
    const float* W_nc    = (const float*)d_in[10];
    const float* b_nc    = (const float*)d_in[11];
    const float* W_des   = (const float*)d_in[12];
    const float* b_des   = (const float*)d_in[13];
    const float* W_text  = (const float*)d_in[14];
    const float* b_text  = (const float*)d_in[15];
    const float* W_tweet = (const float*)d_in[16];
    const float* b_tweet = (const float*)d_in[17];
    const float* W_in    = (const float*)d_in[18];
    const float* b_in    = (const float*)d_in[19];
    const float* W_rel1  = (const float*)d_in[20];
    const float* W_root1 = (const float*)d_in[21];
    const float* b_c1    = (const float*)d_in[22];
    const float* W_rel2  = (const float*)d_in[23];
    const float* W_root2 = (const float*)d_in[24];
    const float* b_c2    = (const float*)d_in[25];
    const float* W_o1    = (const float*)d_in[26];
    const float* b_o1    = (const float*)d_in[27];
    const float* W_o2    = (const float*)d_in[28];
    const float* b_o2    = (const float*)d_in[29];

    const int* src = eidx;
    const int* dst = eidx + N_EDGES;

    // Workspace: h | nh | t | acc | cnt
    const long A = (long)N_NODES * HDIM;
    float* ws  = (float*)d_ws;
    float* h   = ws;
    float* nh  = ws + A;
    float* t   = ws + 2 * A;
    float* acc = ws + 3 * A;
    float* cnt = ws + 4 * A;

    const int gX = (ROWTILES + 3) / 4;       // 782 blocks of 4 row-tiles

    auto gemm = [&](const float* Ap, const float* Wp, int ldw, const float* bp,
                    float* Cp, int ldc, int K, int Nc, int lrelu) {
        dim3 grid(gX, Nc / 16);
        gemm_wmma_f32<<<grid, 128, 0, stream>>>(Ap, Wp, ldw, bp, Cp, ldc, K, lrelu);
    };

    // ---- Feature encoders -> h[N,160] (concat n, c, d, tw, pre_t)
    const int encBlocks = (N_NODES * 32 + 255) / 256;
    encode_small<<<encBlocks, 256, 0, stream>>>(nprop, 6,  W_np, b_np, h + 0);
    encode_small<<<encBlocks, 256, 0, stream>>>(ncat,  11, W_nc, b_nc, h + 32);
    gemm(des,   W_des,   32, b_des,   h +  64, HDIM, 768, 32, 1);
    gemm(tweet, W_text,  32, b_text,  h +  96, HDIM, 768, 32, 1);
    gemm(pre_x, W_tweet, 32, b_tweet, h + 128, HDIM, 768, 32, 1);

    // ---- h = lrelu(h @ W_in + b_in)
    gemm(h, W_in, HDIM, b_in, nh, HDIM, HDIM, HDIM, 1);
    { float* tmp = h; h = nh; nh = tmp; }

    // ---- Two RGCN layers (no inter-layer activation, per reference)
    const float* Wrel[2]  = { W_rel1, W_rel2 };
    const float* Wroot[2] = { W_root1, W_root2 };
    const float* bc[2]    = { b_c1, b_c2 };

    const long eThreads = (long)N_EDGES * (HDIM / 4);
    const int  eBlocks  = (int)((eThreads + 255) / 256);
    const long nThreads = (long)N_NODES * (HDIM / 4);
    const int  nBlocks  = (int)((nThreads + 255) / 256);

    for (int layer = 0; layer < 2; ++layer) {
        gemm(h, Wroot[layer], HDIM, bc[layer], nh, HDIM, HDIM, HDIM, 0);   // root + bias

        for (int r = 0; r < 2; ++r) {
            gemm(h, Wrel[layer] + (long)r * HDIM * HDIM, HDIM,
                 nullptr, t, HDIM, HDIM, HDIM, 0);                          // t = h @ W_rel[r]

            hipMemsetAsync(acc, 0, A * sizeof(float), stream);
            hipMemsetAsync(cnt, 0, N_NODES * sizeof(float), stream);

            scatter_rel4<<<eBlocks, 256, 0, stream>>>(t, src, dst, etype, r, acc, cnt);
            combine_mean4<<<nBlocks, 256, 0, stream>>>(nh, acc, cnt);
        }
        { float* tmp = h; h = nh; nh = tmp; }
    }

    // ---- em = lrelu(h @ W_o1 + b_o1) into d_out tail, then 80->2 head
    float* out_logits = (float*)d_out;
    float* em         = (float*)d_out + (long)N_NODES * 2;
    gemm(h, W_o1, 80, b_o1, em, 80, HDIM, 80, 1);
    out_head<<<(N_NODES + 255) / 256, 256, 0, stream>>>(em, W_o2, b_o2, out_logits);
}